// Attention2_6030134083714
// MI455X (gfx1250) — compile-verified
//
#include <hip/hip_runtime.h>
#include <math.h>

// ---------------- problem constants ----------------
#define HW   2304              // 48*48
#define NB   2
#define NH   8
#define KD   16
#define DV   64
#define DHID 512               // NH*DV
#define CIN  256
#define NT   (NB*HW)           // 4608 tokens
#define NHKD 128               // NH*KD
#define KROW 18                // padded LDS row stride (floats) for K panel

typedef __attribute__((ext_vector_type(2))) float        v2f;
typedef __attribute__((ext_vector_type(8))) float        v8f;
typedef __attribute__((ext_vector_type(4))) unsigned int v4u;
typedef __attribute__((ext_vector_type(8))) int          v8i;
typedef __attribute__((ext_vector_type(4))) int          v4i;

// D = A(16x4, fp32) * B(4x16, fp32) + C(16x16 fp32)
// A/B lane layout (ISA 7.12.2, 32-bit): lane l holds row (l&15),
// K elements kc + 2*(l>>4) + {0,1}. C/D: col = lane&15, row = vgpr + 8*(lane>>4).
__device__ __forceinline__ v8f wmma4(v2f a, v2f b, v8f c) {
  return __builtin_amdgcn_wmma_f32_16x16x4_f32(false, a, false, b, (short)0, c,
                                               false, false);
}

// TDM: DMA a 2304x16 fp32 panel (row-major, contiguous) from global into LDS,
// padding the LDS destination by 2 DWORDs every 16 DWORDs (row stride -> 18
// floats, bank-conflict-free, keeps 8B alignment for ds_load_b64).
__device__ __forceinline__ void tdm_load_k_panel(unsigned lds_addr,
                                                 unsigned long long ga) {
  // D# group 0: count=1 | lds_addr | global_addr | type=2
  v4u g0;
  g0.x = 1u;                                  // count=1, user mode
  g0.y = lds_addr;                            // LDS byte address
  g0.z = (unsigned)(ga & 0xffffffffull);      // global_addr[31:0]
  g0.w = (unsigned)((ga >> 32) & 0x1ffffffull) | (2u << 30);  // [56:32] | type=2
  // D# group 1: data_size=4B (2), pad_enable, pad_interval=3 (16 DW),
  //             pad_amount=1 (2 DW); tensor 16 x 2304, tile 16 x 2304, stride0=16
  v8i g1;
  g1[0] = (int)((2u << 16) | (1u << 20) | (3u << 22) | (1u << 25));
  g1[1] = (int)(16u << 16);                   // tensor_dim0[15:0]=16
  g1[2] = (int)(2304u << 16);                 // tensor_dim0 hi=0 | tensor_dim1 lo
  g1[3] = (int)(16u << 16);                   // tensor_dim1 hi=0 | tile_dim0=16
  g1[4] = (int)2304;                          // tile_dim1=2304, tile_dim2=0
  g1[5] = 16;                                 // tensor_dim0_stride lo
  g1[6] = 0;                                  // stride0 hi | stride1 lo
  g1[7] = 0;                                  // stride1 hi
  v4i z4 = {0, 0, 0, 0};                      // groups 2/3 unused (<=2D tensor)
#if defined(__clang_major__) && __clang_major__ >= 23
  v8i z8 = {0, 0, 0, 0, 0, 0, 0, 0};
  __builtin_amdgcn_tensor_load_to_lds(g0, g1, z4, z4, z8, 0);
#else
  __builtin_amdgcn_tensor_load_to_lds(g0, g1, z4, z4, 0);
#endif
}

// ---------------- kernel 1: fused QKV projection GEMM ----------------
// lin = xf(4608x256) @ W^T + bias for W in {Wq,Wk,Wv}; one wave per 16x16 tile.
__global__ void __launch_bounds__(256) k_qkv(
    const float* __restrict__ x,
    const float* __restrict__ Wq, const float* __restrict__ bq,
    const float* __restrict__ Wk, const float* __restrict__ bk,
    const float* __restrict__ Wv, const float* __restrict__ bv,
    float* __restrict__ linq, float* __restrict__ link,
    float* __restrict__ linv) {
  int wave = (blockIdx.x * blockDim.x + threadIdx.x) >> 5;
  int lane = threadIdx.x & 31;
  if (wave >= (NT / 16) * (768 / 16)) return;
  int half = lane >> 4, lr = lane & 15;
  int mt = wave / 48, nt = wave % 48;
  int m0 = mt * 16, n0g = nt * 16;
  const float* W; const float* bias; float* out; int n0, Nout;
  if (n0g < 128)      { W = Wq; bias = bq; out = linq; n0 = n0g;       Nout = 128; }
  else if (n0g < 256) { W = Wk; bias = bk; out = link; n0 = n0g - 128; Nout = 128; }
  else                { W = Wv; bias = bv; out = linv; n0 = n0g - 256; Nout = 512; }
  // x is (B,C,H,W) but xf = raw view as (B*HW, 256) -> x flat is already (4608,256)
  const float* arow = x + (size_t)(m0 + lr) * CIN + 2 * half;
  const float* brow = W + (size_t)(n0 + lr) * CIN + 2 * half;
  v8f acc = {};
  for (int kc = 0; kc < CIN; kc += 4)
    acc = wmma4(*(const v2f*)(arow + kc), *(const v2f*)(brow + kc), acc);
  float bb = bias[n0 + lr];
#pragma unroll
  for (int r = 0; r < 8; r++) {
    int mrow = m0 + r + 8 * half;
    out[(size_t)mrow * Nout + n0 + lr] = acc[r] + bb;
  }
}

// ---------------- kernel 2: focused feature map for q,k + K^T scatter -------
// fq[b,h,m,k] = linq_flat[b][h*KD*HW + k*HW + m]   (raw reshape quirk)
// focusing: relu+1e-6, /softplus(scale), x^3, renormalize to original L2 norm.
// fk then raw-reshaped: fkr[k][m] = fk[k*144 + m/16][m%16]; we store kT = fkr^T.
__global__ void k_focus(const float* __restrict__ linq,
                        const float* __restrict__ link,
                        const float* __restrict__ scale,
                        float* __restrict__ qf, float* __restrict__ kT) {
  int idx = blockIdx.x * blockDim.x + threadIdx.x;
  if (idx >= NB * NH * HW) return;
  int r = idx % HW;
  int h = (idx / HW) % NH;
  int b = idx / (HW * NH);
  float vq[KD], vk[KD];
  float sq = 0.f, sk = 0.f, s3q = 0.f, s3k = 0.f;
#pragma unroll
  for (int k = 0; k < KD; k++) {
    int fi = h * (KD * HW) + k * HW + r;
    float q  = linq[(size_t)b * (HW * NHKD) + fi];
    float kk = link[(size_t)b * (HW * NHKD) + fi];
    float sp = log1pf(expf(scale[k]));       // softplus
    q  = (fmaxf(q, 0.f) + 1e-6f) / sp;
    kk = (fmaxf(kk, 0.f) + 1e-6f) / sp;
    sq += q * q;  sk += kk * kk;
    q = q * q * q;  kk = kk * kk * kk;
    s3q += q * q;  s3k += kk * kk;
    vq[k] = q;  vk[k] = kk;
  }
  float fqs = sqrtf(sq) * rsqrtf(s3q);
  float fks = sqrtf(sk) * rsqrtf(s3k);
  int base = (b * NH + h) * HW;
  int kcol = r / 144;                 // 144 = HW/16
  int mbase = (r % 144) * 16;
#pragma unroll
  for (int k = 0; k < KD; k++) {
    qf[(size_t)(base + r) * KD + k] = vq[k] * fqs;
    kT[(size_t)(base + mbase + k) * KD + kcol] = vk[k] * fks;
  }
}

// ---------------- kernel 3: build V (B,h,HW,64) and V^T (B,h,64,HW) ---------
// fv[b,h,m,d] = linv_flat[b][h*DV*HW + d*HW + m]  (raw reshape quirk)
__global__ void k_buildv(const float* __restrict__ linv,
                         float* __restrict__ fv, float* __restrict__ fvT) {
  int idx = blockIdx.x * blockDim.x + threadIdx.x;
  if (idx >= NB * NH * HW * DV) return;
  int d = idx % DV;
  int m = (idx / DV) % HW;
  int h = (idx / (DV * HW)) % NH;
  int b = idx / (DV * HW * NH);
  int fi = h * (DV * HW) + d * HW + m;
  float v = linv[(size_t)b * (HW * DHID) + fi];
  int bh = b * NH + h;
  fv [((size_t)bh * HW + m) * DV + d] = v;
  fvT[((size_t)bh * DV + d) * HW + m] = v;
}

// ---------------- kernel 4: 5x5 depthwise conv over the (HW,64) plane -------
__global__ void k_dwc(const float* __restrict__ fv, const float* __restrict__ w,
                      const float* __restrict__ bdw, float* __restrict__ fmb) {
  int idx = blockIdx.x * blockDim.x + threadIdx.x;
  if (idx >= NB * NH * HW * DV) return;
  int d = idx % DV;
  int m = (idx / DV) % HW;
  int h = (idx / (DV * HW)) % NH;
  int b = idx / (DV * HW * NH);
  int bh = b * NH + h;
  const float* plane = fv + (size_t)bh * HW * DV;
  float acc = bdw[h];
#pragma unroll
  for (int i = 0; i < 5; i++) {
    int mm = m + i - 2;
    if (mm < 0 || mm >= HW) continue;
#pragma unroll
    for (int j = 0; j < 5; j++) {
      int dd = d + j - 2;
      if (dd < 0 || dd >= DV) continue;
      acc += plane[(size_t)mm * DV + dd] * w[h * 25 + i * 5 + j];
    }
  }
  fmb[((size_t)bh * HW + m) * DV + d] = acc;
}

// ---------------- kernel 5: fused two-pass softmax attention ----------------
// 4 waves/block, each owning a 16-row query tile of one (b,h).
// The block's whole K^T panel (2304x16 f32) is DMA'd into LDS once by the TDM
// (padded rows, stride 18 floats). Pass 1: S = Q K^T via wmma4, row max.
// Pass 2: recompute S, P = exp(S-max), row-sum, LDS-transpose P to A layout,
// O += P*V via wmma4 against V^T; epilogue: O/rowsum + dwc, ReLU, store.
__global__ void __launch_bounds__(128) k_attn(
    const float* __restrict__ qf, const float* __restrict__ kT,
    const float* __restrict__ fvT, const float* __restrict__ fmb,
    float* __restrict__ xxr) {
  extern __shared__ float smem[];
  float* sK = smem;                     // HW * KROW floats (padded K panel)
  float* sP = smem + HW * KROW;         // 4 * 16 * 17 floats (P transpose)
  int w = threadIdx.x >> 5;
  int lane = threadIdx.x & 31;
  int half = lane >> 4, lr = lane & 15;
  int bh = blockIdx.x / 36;
  int g  = blockIdx.x % 36;
  int n0 = (g * 4 + w) * 16;
  const float* Q  = qf  + (size_t)bh * HW * KD;
  const float* VT = fvT + (size_t)bh * DV * HW;
  const float* qrow = Q + (size_t)(n0 + lr) * KD + 2 * half;

  // one wave launches the TDM copy of this (b,h)'s K^T panel into LDS
  if (w == 0) {
    tdm_load_k_panel((unsigned)(size_t)(void*)sK,
                     (unsigned long long)(size_t)(kT + (size_t)bh * HW * KD));
    __builtin_amdgcn_s_wait_tensorcnt(0);
  }
  __syncthreads();

  float rmax[8];
#pragma unroll
  for (int r = 0; r < 8; r++) rmax[r] = -3.0e38f;

  for (int m0 = 0; m0 < HW; m0 += 16) {
    const float* krow = sK + (size_t)(m0 + lr) * KROW + 2 * half;
    v8f s = {};
#pragma unroll
    for (int kc = 0; kc < KD; kc += 4)
      s = wmma4(*(const v2f*)(qrow + kc), *(const v2f*)(krow + kc), s);
#pragma unroll
    for (int r = 0; r < 8; r++) {
      float v = s[r];
      v = fmaxf(v, __shfl_xor(v, 1, 32));
      v = fmaxf(v, __shfl_xor(v, 2, 32));
      v = fmaxf(v, __shfl_xor(v, 4, 32));
      v = fmaxf(v, __shfl_xor(v, 8, 32));
      rmax[r] = fmaxf(rmax[r], v);
    }
  }

  float rsum[8];
#pragma unroll
  for (int r = 0; r < 8; r++) rsum[r] = 0.f;
  v8f o[4] = {};

  for (int m0 = 0; m0 < HW; m0 += 16) {
    const float* krow = sK + (size_t)(m0 + lr) * KROW + 2 * half;
    __builtin_prefetch(VT + (size_t)lr * HW + m0 + 16, 0, 1);
    v8f s = {};
#pragma unroll
    for (int kc = 0; kc < KD; kc += 4)
      s = wmma4(*(const v2f*)(qrow + kc), *(const v2f*)(krow + kc), s);
#pragma unroll
    for (int r = 0; r < 8; r++) {
      float p = expf(s[r] - rmax[r]);
      s[r] = p;
      float t = p;
      t += __shfl_xor(t, 1, 32);
      t += __shfl_xor(t, 2, 32);
      t += __shfl_xor(t, 4, 32);
      t += __shfl_xor(t, 8, 32);
      rsum[r] += t;
    }
    __syncthreads();
#pragma unroll
    for (int r = 0; r < 8; r++)                     // store P (D layout)
      sP[(w * 16 + r + 8 * half) * 17 + lr] = s[r];
    __syncthreads();
#pragma unroll
    for (int mc = 0; mc < 16; mc += 4) {            // reload P (A layout)
      v2f a;
      a.x = sP[(w * 16 + lr) * 17 + mc + 2 * half];
      a.y = sP[(w * 16 + lr) * 17 + mc + 2 * half + 1];
      int mm = m0 + mc + 2 * half;
#pragma unroll
      for (int db = 0; db < 4; db++) {
        v2f bvec = *(const v2f*)(VT + (size_t)(db * 16 + lr) * HW + mm);
        o[db] = wmma4(a, bvec, o[db]);
      }
    }
  }

#pragma unroll
  for (int db = 0; db < 4; db++) {
#pragma unroll
    for (int r = 0; r < 8; r++) {
      int n = n0 + r + 8 * half;
      int d = db * 16 + lr;
      size_t p = ((size_t)bh * HW + n) * DV + d;
      float val = o[db][r] / rsum[r] + fmb[p];
      xxr[p] = fmaxf(val, 0.f);                     // ReLU feeding proj
    }
  }
}

// ---------------- kernel 6: 1x1 projection GEMM + BatchNorm -----------------
// y[b,o,hw] = BN( sum_c proj_w[o,c] * relu(xx)[b,c,hw] ), c = h*64+d,
// relu(xx) already stored as xxr (B,h,HW,64) -> B-operand rows are hw.
__global__ void __launch_bounds__(256) k_proj(
    const float* __restrict__ xxr, const float* __restrict__ pw,
    const float* __restrict__ gamma, const float* __restrict__ beta,
    const float* __restrict__ mean, const float* __restrict__ var,
    float* __restrict__ out) {
  int wave = (blockIdx.x * blockDim.x + threadIdx.x) >> 5;
  int lane = threadIdx.x & 31;
  if (wave >= NB * 16 * 144) return;
  int half = lane >> 4, lr = lane & 15;
  int b = wave / (16 * 144);
  int t = wave % (16 * 144);
  int o0  = (t / 144) * 16;
  int hw0 = (t % 144) * 16;
  const float* arow = pw + (size_t)(o0 + lr) * DHID + 2 * half;
  int hw = hw0 + lr;
  v8f acc = {};
  for (int kc = 0; kc < DHID; kc += 4) {
    int c = kc + 2 * half;                       // c even -> c,c+1 in same head
    const float* brow =
        xxr + ((size_t)(b * NH + (c >> 6)) * HW + hw) * DV + (c & 63);
    acc = wmma4(*(const v2f*)(arow + kc), *(const v2f*)brow, acc);
  }
#pragma unroll
  for (int r = 0; r < 8; r++) {
    int oc = o0 + r + 8 * half;
    float inv = gamma[oc] * rsqrtf(var[oc] + 1e-5f);
    out[((size_t)b * CIN + oc) * HW + hw] = (acc[r] - mean[oc]) * inv + beta[oc];
  }
}

// ---------------- host launcher ----------------
extern "C" void kernel_launch(void* const* d_in, const int* in_sizes, int n_in,
                              void* d_out, int out_size, void* d_ws,
                              size_t ws_size, hipStream_t stream) {
  const float* x     = (const float*)d_in[0];
  const float* Wq    = (const float*)d_in[1];
  const float* bq    = (const float*)d_in[2];
  const float* Wk    = (const float*)d_in[3];
  const float* bk    = (const float*)d_in[4];
  const float* Wv    = (const float*)d_in[5];
  const float* bv    = (const float*)d_in[6];
  const float* scale = (const float*)d_in[7];
  const float* dwcw  = (const float*)d_in[8];
  const float* dwcb  = (const float*)d_in[9];
  const float* projw = (const float*)d_in[10];
  const float* gamma = (const float*)d_in[11];
  const float* beta  = (const float*)d_in[12];
  const float* mean  = (const float*)d_in[13];
  const float* var   = (const float*)d_in[14];

  float* ws = (float*)d_ws;
  size_t off = 0;
  float* linq = ws + off; off += (size_t)NT * NHKD;          // 589824
  float* link = ws + off; off += (size_t)NT * NHKD;
  float* linv = ws + off; off += (size_t)NT * DHID;          // 2359296
  float* qfb  = ws + off; off += (size_t)NB * NH * HW * KD;
  float* kTb  = ws + off; off += (size_t)NB * NH * HW * KD;
  float* fvb  = ws + off; off += (size_t)NB * NH * HW * DV;
  float* fvTb = ws + off; off += (size_t)NB * NH * HW * DV;
  float* fmb  = ws + off; off += (size_t)NB * NH * HW * DV;
  float* xxrb = ws + off; off += (size_t)NB * NH * HW * DV;

  // 1) QKV projection: 288*48 = 13824 tiles, 8 waves per 256-thread block
  k_qkv<<<1728, 256, 0, stream>>>(x, Wq, bq, Wk, bk, Wv, bv, linq, link, linv);
  // 2) focusing nonlinearity + K^T scatter (one thread per (b,h,row))
  k_focus<<<(NB * NH * HW + 255) / 256, 256, 0, stream>>>(linq, link, scale,
                                                          qfb, kTb);
  // 3) V / V^T layout build
  k_buildv<<<(NB * NH * HW * DV) / 256, 256, 0, stream>>>(linv, fvb, fvTb);
  // 4) depthwise 5x5 conv over (HW, 64) plane
  k_dwc<<<(NB * NH * HW * DV) / 256, 256, 0, stream>>>(fvb, dwcw, dwcb, fmb);
  // 5) fused attention: 16 (b,h) * 36 groups of 4 query tiles;
  //    dynamic LDS: padded K panel (2304*18 f32) + P transpose (4*16*17 f32)
  size_t lds_bytes = (size_t)(HW * KROW + 4 * 16 * 17) * sizeof(float);
  k_attn<<<NB * NH * 36, 128, lds_bytes, stream>>>(qfb, kTb, fvTb, fmb, xxrb);
  // 6) projection GEMM + BN: 2*16*144 = 4608 tiles, 8 waves/block
  k_proj<<<576, 256, 0, stream>>>(xxrb, projw, gamma, beta, mean, var,
                                  (float*)d_out);
}